// RNAGNN_13649406067361
// MI455X (gfx1250) — compile-verified
//
#include <hip/hip_runtime.h>

typedef float v2f __attribute__((ext_vector_type(2)));
typedef float v8f __attribute__((ext_vector_type(8)));

#define BN_EPS 1e-5f

// ---------------- degree / normalization ----------------

__global__ void k_deg_init(float* __restrict__ deg, int N) {
  int i = blockIdx.x * blockDim.x + threadIdx.x;
  if (i < N) deg[i] = 1.0f;                       // self-loop contributes 1
}

__global__ void k_deg_accum(const long long* __restrict__ ei, float* __restrict__ deg,
                            long long E) {
  long long e = (long long)blockIdx.x * blockDim.x + threadIdx.x;
  if (e < E) atomicAdd(&deg[(int)ei[E + e]], 1.0f);   // dst = ei[1][e]
}

__global__ void k_deg_rsqrt(float* __restrict__ deg, int N) {
  int i = blockIdx.x * blockDim.x + threadIdx.x;
  if (i < N) deg[i] = rsqrtf(deg[i]);             // deg >= 1 always
}

// ---------------- dense GEMM via V_WMMA_F32_16X16X4_F32 ----------------
// Y[N,16] = X[N,K] @ W[K,16]; one 16-row tile per wave32.

__global__ void k_gemm_wmma(const float* __restrict__ X, const float* __restrict__ W,
                            float* __restrict__ Y, int n_tiles, int K) {
  int wave = (int)(((long long)blockIdx.x * blockDim.x + threadIdx.x) >> 5);
  int lane = threadIdx.x & 31;
  if (wave >= n_tiles) return;                    // wave-uniform: EXEC all-ones below
  const int m = lane & 15;                        // row (A) / col (B)
  const int khalf = (lane >> 4) << 1;             // lanes 0-15: K=0,1; lanes 16-31: K=2,3
  const long long row_base = (long long)wave * 16;

  v8f acc = {};
  for (int k0 = 0; k0 < K; k0 += 4) {
    v2f a = *(const v2f*)(X + (row_base + m) * K + (k0 + khalf));  // 8B aligned
    v2f b;
    b.x = W[(k0 + khalf) * 16 + m];
    b.y = W[(k0 + khalf + 1) * 16 + m];
    acc = __builtin_amdgcn_wmma_f32_16x16x4_f32(
        /*neg_a=*/false, a, /*neg_b=*/false, b,
        /*c_mod=*/(short)0, acc, /*reuse_a=*/false, /*reuse_b=*/false);
  }
  // C/D layout: VGPR r -> M=r (lanes 0-15) / M=r+8 (lanes 16-31), N=lane&15
  const long long row = row_base + ((lane >> 4) << 3);
#pragma unroll
  for (int r = 0; r < 8; ++r)
    Y[(row + r) * 16 + m] = acc[r];
}

// scalar fallback for a partial tail tile (not used when 16 | N, kept for safety)
__global__ void k_gemm_tail(const float* __restrict__ X, const float* __restrict__ W,
                            float* __restrict__ Y, int row0, int N, int K) {
  int t = blockIdx.x * blockDim.x + threadIdx.x;
  int nrem = (N - row0) * 16;
  if (t >= nrem) return;
  int r = row0 + (t >> 4), c = t & 15;
  float s = 0.f;
  for (int k = 0; k < K; ++k) s += X[(long long)r * K + k] * W[k * 16 + c];
  Y[(long long)r * 16 + c] = s;
}

// ---------------- aggregation: self-loop + bias init, then edge scatter ----------------

__global__ void k_agg_init(const float* __restrict__ XW, const float* __restrict__ dinv,
                           const float* __restrict__ bias, float* __restrict__ OUT,
                           long long NE) {
  long long t = (long long)blockIdx.x * blockDim.x + threadIdx.x;
  if (t >= NE) return;
  int i = (int)(t >> 4), c = (int)(t & 15);
  float di = dinv[i];
  OUT[t] = bias[c] + di * di * XW[t];
}

// 4 threads per edge, 4 channels each (float4 gather + 4 f32 atomics)
__global__ void k_scatter(const long long* __restrict__ ei, const float* __restrict__ dinv,
                          const float* __restrict__ XW, float* __restrict__ OUT,
                          long long E) {
  long long t = (long long)blockIdx.x * blockDim.x + threadIdx.x;
  if (t >= E * 4) return;
  long long e = t >> 2;
  int g = (int)(t & 3);
  int s = (int)ei[e];
  int d = (int)ei[E + e];
  float nrm = dinv[s] * dinv[d];
  float4 v = ((const float4*)(XW + (long long)s * 16))[g];
  float* o = OUT + (long long)d * 16 + g * 4;
  atomicAdd(o + 0, nrm * v.x);
  atomicAdd(o + 1, nrm * v.y);
  atomicAdd(o + 2, nrm * v.z);
  atomicAdd(o + 3, nrm * v.w);
}

// ---------------- BatchNorm (training mode, biased var) + ReLU ----------------

__global__ void k_zero_stats(float* __restrict__ stats) {
  if (threadIdx.x < 32) stats[threadIdx.x] = 0.0f;
}

__global__ void k_bn_stats(const float* __restrict__ H, float* __restrict__ stats, int N) {
  __shared__ float ssum[256];
  __shared__ float ssq[256];
  int t = threadIdx.x;
  int c = t & 15;          // channel
  int grp = t >> 4;        // 16 row-groups per block
  float sum = 0.f, sq = 0.f;
  for (long long i = (long long)blockIdx.x * 16 + grp; i < N; i += (long long)gridDim.x * 16) {
    float v = H[i * 16 + c];
    sum += v;
    sq += v * v;
  }
  ssum[t] = sum; ssq[t] = sq;
  __syncthreads();
  for (int s = 128; s >= 16; s >>= 1) {       // strides are multiples of 16: channel-safe
    if (t < s) { ssum[t] += ssum[t + s]; ssq[t] += ssq[t + s]; }
    __syncthreads();
  }
  if (t < 16) {
    atomicAdd(&stats[c], ssum[t]);
    atomicAdd(&stats[16 + c], ssq[t]);
  }
}

__global__ void k_bn_apply(float* __restrict__ H, const float* __restrict__ stats,
                           const float* __restrict__ gamma, const float* __restrict__ beta,
                           long long NE, float invN) {
  long long t = (long long)blockIdx.x * blockDim.x + threadIdx.x;
  if (t >= NE) return;
  int c = (int)(t & 15);
  float mean = stats[c] * invN;
  float var = stats[16 + c] * invN - mean * mean;
  float scale = gamma[c] * rsqrtf(var + BN_EPS);
  float v = (H[t] - mean) * scale + beta[c];
  H[t] = fmaxf(v, 0.0f);
}

__global__ void k_relu(float* __restrict__ p, long long n) {
  long long t = (long long)blockIdx.x * blockDim.x + threadIdx.x;
  if (t < n) p[t] = fmaxf(p[t], 0.0f);
}

// ---------------- host ----------------

extern "C" void kernel_launch(void* const* d_in, const int* in_sizes, int n_in,
                              void* d_out, int out_size, void* d_ws, size_t ws_size,
                              hipStream_t stream) {
  const float*     x      = (const float*)d_in[0];
  const long long* ei     = (const long long*)d_in[1];
  const float*     W_in   = (const float*)d_in[2];
  const float*     b_in   = (const float*)d_in[3];
  const float*     Ws     = (const float*)d_in[4];
  const float*     bs     = (const float*)d_in[5];
  const float*     gammas = (const float*)d_in[6];
  const float*     betas  = (const float*)d_in[7];
  const float*     W_sin  = (const float*)d_in[8];
  const float*     b_sin  = (const float*)d_in[9];
  const float*     W_cos  = (const float*)d_in[10];
  const float*     b_cos  = (const float*)d_in[11];

  const int       DIM  = in_sizes[3];                 // 16
  const int       F_IN = in_sizes[2] / DIM;           // 64
  const int       N    = in_sizes[0] / F_IN;          // 200000
  const long long E    = (long long)in_sizes[1] / 2;  // 6400000
  const int       L    = in_sizes[4] / (DIM * DIM);   // 4

  float* dinv  = (float*)d_ws;                        // [N]
  float* bufA  = dinv + N;                            // [N,16] xw scratch
  float* bufB  = bufA + (long long)N * 16;            // [N,16] h
  float* stats = bufB + (long long)N * 16;            // [32]
  float* outs  = (float*)d_out;                       // [2,N,16]

  const long long NE = (long long)N * 16;
  const int blk = 256;
  const float invN = 1.0f / (float)N;

  // degree -> dinv
  k_deg_init<<<(N + blk - 1) / blk, blk, 0, stream>>>(dinv, N);
  k_deg_accum<<<(int)((E + blk - 1) / blk), blk, 0, stream>>>(ei, dinv, E);
  k_deg_rsqrt<<<(N + blk - 1) / blk, blk, 0, stream>>>(dinv, N);

  auto conv = [&](const float* Hin, int K, const float* W, const float* b, float* out) {
    int n_tiles = N / 16;
    int waves_per_blk = blk / 32;
    int gblocks = (n_tiles + waves_per_blk - 1) / waves_per_blk;
    k_gemm_wmma<<<gblocks, blk, 0, stream>>>(Hin, W, bufA, n_tiles, K);
    int rem_rows = N - n_tiles * 16;
    if (rem_rows > 0)
      k_gemm_tail<<<(rem_rows * 16 + blk - 1) / blk, blk, 0, stream>>>(Hin, W, bufA,
                                                                       n_tiles * 16, N, K);
    k_agg_init<<<(int)((NE + blk - 1) / blk), blk, 0, stream>>>(bufA, dinv, b, out, NE);
    k_scatter<<<(int)((E * 4 + blk - 1) / blk), blk, 0, stream>>>(ei, dinv, bufA, out, E);
  };

  // input conv: h = gcn(x, W_in, b_in)   (no BN/ReLU per reference)
  conv(x, F_IN, W_in, b_in, bufB);

  // hidden layers: h = relu(bn(gcn(h, Ws[l], bs[l])))
  for (int l = 0; l < L; ++l) {
    conv(bufB, DIM, Ws + (long long)l * DIM * DIM, bs + (long long)l * DIM, bufB);
    k_zero_stats<<<1, 32, 0, stream>>>(stats);
    k_bn_stats<<<256, 256, 0, stream>>>(bufB, stats, N);
    k_bn_apply<<<(int)((NE + blk - 1) / blk), blk, 0, stream>>>(
        bufB, stats, gammas + (long long)l * DIM, betas + (long long)l * DIM, NE, invN);
  }

  // outputs: relu(gcn(h, W_sin/b_sin)), relu(gcn(h, W_cos/b_cos))
  conv(bufB, DIM, W_sin, b_sin, outs);
  conv(bufB, DIM, W_cos, b_cos, outs + NE);
  k_relu<<<(int)((2 * NE + blk - 1) / blk), blk, 0, stream>>>(outs, 2 * NE);
}